// AlighQuantizer_9294309229201
// MI455X (gfx1250) — compile-verified
//
#include <hip/hip_runtime.h>
#include <stdint.h>

// AlignQuantizer for MI455X (gfx1250, wave32).
// Per group of 256: m = floor(log2(max|x| + 1e-30));
//                   out = trunc(x * 2^(10-m)) * 2^(m-10)     (== sign*floor(|x|*s)*inv)
// Memory-bound: 512 MB total traffic -> ~22 us roofline @ 23.3 TB/s.

#define GS  256
#define WPB 8            // waves per block (256 threads, wave32)

typedef float v4f __attribute__((ext_vector_type(4)));
typedef int   v4i __attribute__((ext_vector_type(4)));
typedef __attribute__((address_space(1))) v4i gv4i;  // global int4
typedef __attribute__((address_space(3))) v4i lv4i;  // LDS int4

#if defined(__gfx1250__) && __has_builtin(__builtin_amdgcn_global_load_async_to_lds_b128)
#define USE_ASYNC 1
#else
#define USE_ASYNC 0
#endif

template <int N>
__device__ __forceinline__ void wait_async_le() {
#if __has_builtin(__builtin_amdgcn_s_wait_asynccnt)
    __builtin_amdgcn_s_wait_asynccnt(N);
#else
    asm volatile("s_wait_asynccnt %0" ::"n"(N) : "memory");
#endif
}

// Full-wave32 xor-swap max reduction: ds_swizzle group-of-32 mode
// (offset = xor_mask<<10 | and_mask 0x1f), 1 DS + 1 VALU per step.
__device__ __forceinline__ float wave_max32(float v) {
    int b = __float_as_int(v);
#define SWZ(mask)                                                                        \
    b = __float_as_int(fmaxf(__int_as_float(b),                                          \
            __int_as_float(__builtin_amdgcn_ds_swizzle(b, ((mask) << 10) | 0x1f))));
    SWZ(16) SWZ(8) SWZ(4) SWZ(2) SWZ(1)
#undef SWZ
    return __int_as_float(b);
}

__device__ __forceinline__ float lane_max8(v4f a, v4f b) {
    // |src| modifiers fold into v_max_f32 for free
    return fmaxf(fmaxf(fmaxf(fabsf(a.x), fabsf(a.y)), fmaxf(fabsf(a.z), fabsf(a.w))),
                 fmaxf(fmaxf(fabsf(b.x), fabsf(b.y)), fmaxf(fabsf(b.z), fabsf(b.w))));
}

__device__ __forceinline__ void quant_store(v4f a, v4f b, float* __restrict__ outp, int lane) {
    // group max of |x| (max commutes with +eps and floor(log2(.)))
    const float m  = wave_max32(lane_max8(a, b));
    const float y  = m + 1e-30f;                    // y >= 1e-30 -> always normal
    const int   mi = (int)floorf(__log2f(y));       // v_log_f32 + v_floor, matches ref semantics
    const float s   = __int_as_float((137 - mi) << 23);  // 2^(10 - mi), exp field in [9,237]
    const float inv = __int_as_float((117 + mi) << 23);  // 2^(mi - 10), exp field in [17,244]

    // sign(x)*floor(|x|*s)*inv == trunc(x*s)*inv  (trunc toward zero; ±0 preserved)
    v4f ra, rb;
    ra.x = truncf(a.x * s) * inv;
    ra.y = truncf(a.y * s) * inv;
    ra.z = truncf(a.z * s) * inv;
    ra.w = truncf(a.w * s) * inv;
    rb.x = truncf(b.x * s) * inv;
    rb.y = truncf(b.y * s) * inv;
    rb.z = truncf(b.z * s) * inv;
    rb.w = truncf(b.w * s) * inv;

    // streaming stores: don't pollute the 192MB L2 with write-once data
    __builtin_nontemporal_store(ra, (v4f*)(outp + 4 * lane));
    __builtin_nontemporal_store(rb, (v4f*)(outp + 128 + 4 * lane));
}

__global__ __launch_bounds__(WPB * 32) void align_quant_kernel(const float* __restrict__ x,
                                                               float* __restrict__ out,
                                                               int num_groups, int total_waves) {
    const int lane = threadIdx.x & 31;
    const int w    = threadIdx.x >> 5;
    int g = blockIdx.x * WPB + w;   // one wave32 per 256-element group

#if USE_ASYNC
    // Double-buffered async global->LDS pipeline per wave (ASYNCcnt-tracked,
    // no VGPR pressure from in-flight loads). 2 x b128 per lane per group:
    // each async b128 moves a contiguous 512B per wave.
    __shared__ v4f lds[2][WPB][64];   // 16 KB / block

    if (g >= num_groups) return;      // wave-uniform: EXEC stays all-ones for survivors

    auto issue = [&](int gg, int bufi) {
        const float* src = x + (size_t)gg * GS + 4 * lane;
        __builtin_amdgcn_global_load_async_to_lds_b128(
            (gv4i*)(void*)src, (lv4i*)&lds[bufi][w][lane], 0, 0);
        __builtin_amdgcn_global_load_async_to_lds_b128(
            (gv4i*)(void*)(src + 128), (lv4i*)&lds[bufi][w][32 + lane], 0, 0);
    };

    issue(g, 0);
    int buf = 0;
    for (;;) {
        const int  gn   = g + total_waves;
        const bool more = gn < num_groups;
        if (more) {
            issue(gn, buf ^ 1);
            wait_async_le<2>();   // current buffer's 2 loads complete (in-order)
        } else {
            wait_async_le<0>();
        }
        v4f a = lds[buf][w][lane];
        v4f b = lds[buf][w][32 + lane];
        quant_store(a, b, out + (size_t)g * GS, lane);
        if (!more) break;
        g = gn;
        buf ^= 1;
    }
#else
    for (; g < num_groups; g += total_waves) {
        const float* src = x + (size_t)g * GS + 4 * lane;
        v4f a = __builtin_nontemporal_load((const v4f*)src);
        v4f b = __builtin_nontemporal_load((const v4f*)(src + 128));
        quant_store(a, b, out + (size_t)g * GS, lane);
    }
#endif
}

extern "C" void kernel_launch(void* const* d_in, const int* in_sizes, int n_in,
                              void* d_out, int out_size, void* d_ws, size_t ws_size,
                              hipStream_t stream) {
    (void)n_in; (void)out_size; (void)d_ws; (void)ws_size;
    const float* x   = (const float*)d_in[0];
    float*       out = (float*)d_out;

    const long long n          = in_sizes[0];
    const int       num_groups = (int)(n / GS);

    int blocks = (num_groups + WPB - 1) / WPB;
    if (blocks > 8192) blocks = 8192;        // grid-stride: ~4 groups/wave -> pipeline depth
    const int total_waves = blocks * WPB;

    hipLaunchKernelGGL(align_quant_kernel, dim3(blocks), dim3(WPB * 32), 0, stream,
                       x, out, num_groups, total_waves);
}